// PillarFeatureNet_10359461118353
// MI455X (gfx1250) — compile-verified
//
#include <hip/hip_runtime.h>

// ---------------------------------------------------------------------------
// PillarFeatureNet forward for MI455X (gfx1250), wave32.
//   Matrix op: v_wmma_f32_16x16x4_f32 (f32 in/out, exact vs. reference).
//   Data movement: global_load_async_to_lds_b128 double-buffered staging,
//   drained with s_wait_asynccnt (ASYNCcnt path).
//   Pass 1 (k_stats):  per-pillar 64x64xK12 GEMM, register-accumulated
//                      sum/sumsq per point-index n, deterministic partials.
//   Pass 2 (k_params): fold partials into affine BN params a[n], b[n].
//   Pass 3 (k_out):    recompute GEMM, y = a[n]*x + b[n], relu+max over n.
// ---------------------------------------------------------------------------

typedef __attribute__((ext_vector_type(2))) float v2f;
typedef __attribute__((ext_vector_type(8))) float v8f;

#define P_TOT   30000
#define NPT     64
#define CIN     4
#define OUTC    64
#define KPAD    12                // C_AUG = 9 padded to 12 (3 chunks of K=4)
#define SBLK    240               // stats blocks; 30000 / 240 = 125 exactly
#define PPB_S   (P_TOT / SBLK)    // pillars per stats block
#define OBLK    3750              // output blocks; 30000 / 3750 = 8 exactly
#define PPB_O   (P_TOT / OBLK)    // pillars per output block
#define PAR_OFF (SBLK * 128)      // float offset of a[]/b[] params in d_ws
#define COUNT_F 1920000.0f        // P * OUT  (BN count per n)
#define BN_EPS  0.001f
#define C_VX    0.2f
#define C_VY    0.2f
#define C_XOFF  (0.5f * C_VX + 0.0f)
#define C_YOFF  (0.5f * C_VY - 40.0f)

#define USE_ASYNC 1   // gfx1250 async global->LDS engine; 0 = sync fallback

// Copy 16B from global to LDS via the async engine (ASYNCcnt-tracked).
// LDS destination address = low 32 bits of the generic pointer.
__device__ __forceinline__ void async_stage16(void* lds_dst, const void* gsrc) {
#if USE_ASYNC
  const unsigned lds_off = (unsigned)(unsigned long long)lds_dst;
  asm volatile("global_load_async_to_lds_b128 %0, %1, off"
               :: "v"(lds_off), "v"(gsrc)
               : "memory");
#else
  *(float4*)lds_dst = *(const float4*)gsrc;
#endif
}

__device__ __forceinline__ void async_wait_le1() {
#if USE_ASYNC
  asm volatile("s_wait_asynccnt 0x1" ::: "memory");
#endif
}
__device__ __forceinline__ void async_wait_le0() {
#if USE_ASYNC
  asm volatile("s_wait_asynccnt 0x0" ::: "memory");
#endif
}

// Load W (OUT x 9, row-major) into LDS transposed+padded as Wt[12][64], then
// build per-lane constant B fragments (lane&15 = N column; K split across
// lane halves, symmetric to the documented f32 A layout).
__device__ __forceinline__ void build_bfrags(const float* __restrict__ W,
                                             float (*Wt)[OUTC],
                                             v2f bfrag[3][4]) {
  const int t = threadIdx.x;
  for (int i = t; i < KPAD * OUTC; i += 128) {
    const int k = i / OUTC, o = i % OUTC;
    Wt[k][o] = (k < 9) ? W[o * 9 + k] : 0.f;
  }
  __syncthreads();
  const int lane = t & 31;
#pragma unroll
  for (int kc = 0; kc < 3; ++kc) {
    const int kk = kc * 4 + ((lane >> 4) << 1);
#pragma unroll
    for (int j = 0; j < 4; ++j) {
      const int col = j * 16 + (lane & 15);
      v2f b;
      b.x = Wt[kk][col];
      b.y = Wt[kk + 1][col];
      bfrag[kc][j] = b;
    }
  }
}

// 64x64 GEMM (K padded to 12) for one pillar. Wave w owns rows [16w,16w+16).
// A layout (f32 16x4): lane&15 = M; lanes<16 hold K0/K1, lanes>=16 K2/K3.
__device__ __forceinline__ void pillar_gemm(const float (*aug)[KPAD],
                                            const v2f bfrag[3][4],
                                            v8f cacc[4]) {
  const int lane = threadIdx.x & 31;
  const int w = threadIdx.x >> 5;
  const int row = (w << 4) + (lane & 15);
  const v8f zero = {0.f, 0.f, 0.f, 0.f, 0.f, 0.f, 0.f, 0.f};
#pragma unroll
  for (int j = 0; j < 4; ++j) cacc[j] = zero;
#pragma unroll
  for (int kc = 0; kc < 3; ++kc) {
    const int kk = kc * 4 + ((lane >> 4) << 1);     // even -> 8B aligned
    const v2f a = *(const v2f*)&aug[row][kk];
#pragma unroll
    for (int j = 0; j < 4; ++j) {
      cacc[j] = __builtin_amdgcn_wmma_f32_16x16x4_f32(
          /*neg_a=*/false, a, /*neg_b=*/false, bfrag[kc][j],
          /*c_mod=*/(short)0, cacc[j], /*reuse_a=*/false, /*reuse_b=*/false);
    }
  }
}

// Given staged raw features fb[64][4] in LDS: wave 0 shuffle-reduces the
// unmasked xyz sums (no LDS tree, 1 barrier), then t<64 builds the masked
// 12-channel augmented features, then all 4 waves run the WMMA GEMM.
__device__ __forceinline__ void pillar_compute(const float (*fb)[CIN],
                                               const int* __restrict__ npts,
                                               const int* __restrict__ coors,
                                               int p,
                                               float (*aug)[KPAD],
                                               float* sums,
                                               const v2f bfrag[3][4],
                                               v8f cacc[4]) {
  const int t = threadIdx.x;
  const int lane = t & 31;
  if (t < 32) {
    float sx = fb[lane][0] + fb[lane + 32][0];
    float sy = fb[lane][1] + fb[lane + 32][1];
    float sz = fb[lane][2] + fb[lane + 32][2];
#pragma unroll
    for (int m = 16; m > 0; m >>= 1) {
      sx += __shfl_xor(sx, m, 32);
      sy += __shfl_xor(sy, m, 32);
      sz += __shfl_xor(sz, m, 32);
    }
    if (lane == 0) { sums[0] = sx; sums[1] = sy; sums[2] = sz; }
  }
  __syncthreads();
  if (t < NPT) {
    const float4 f = *(const float4*)fb[t];
    const int np = npts[p];
    const float inv_np = 1.0f / (float)np;
    const float mx = sums[0] * inv_np;
    const float my = sums[1] * inv_np;
    const float mz = sums[2] * inv_np;
    const float cxo = (float)coors[p * 4 + 3] * C_VX + C_XOFF;
    const float cyo = (float)coors[p * 4 + 2] * C_VY + C_YOFF;
    const float m = (t < np) ? 1.0f : 0.0f;
    aug[t][0]  = f.x * m;
    aug[t][1]  = f.y * m;
    aug[t][2]  = f.z * m;
    aug[t][3]  = f.w * m;
    aug[t][4]  = (f.x - mx) * m;
    aug[t][5]  = (f.y - my) * m;
    aug[t][6]  = (f.z - mz) * m;
    aug[t][7]  = (f.x - cxo) * m;
    aug[t][8]  = (f.y - cyo) * m;
    aug[t][9]  = 0.f;
    aug[t][10] = 0.f;
    aug[t][11] = 0.f;
  }
  __syncthreads();
  pillar_gemm(aug, bfrag, cacc);
}

// ---- Pass 1: register-accumulated per-n sum / sumsq partials --------------
__global__ __launch_bounds__(128) void k_stats(const float* __restrict__ feats,
                                               const int* __restrict__ npts,
                                               const int* __restrict__ coors,
                                               const float* __restrict__ W,
                                               float* __restrict__ ws) {
  __shared__ __align__(16) float fbuf[2][NPT][CIN];
  __shared__ float aug[NPT][KPAD];
  __shared__ float Wt[KPAD][OUTC];
  __shared__ float sums[4];

  const int t = threadIdx.x;
  const int lane = t & 31;
  const int w = t >> 5;
  v2f bfrag[3][4];
  build_bfrags(W, Wt, bfrag);   // includes a barrier

  // Per-lane accumulators: slot v maps to fixed n = 16w+v+8*(lane>=16),
  // fixed o-column = 16j+(lane&15); accumulate over all pillars in registers.
  float sacc[8], qacc[8];
#pragma unroll
  for (int v = 0; v < 8; ++v) { sacc[v] = 0.f; qacc[v] = 0.f; }

  const int p0 = (int)blockIdx.x * PPB_S;
  if (t < NPT)
    async_stage16(&fbuf[0][t][0], feats + ((size_t)p0 * NPT + t) * CIN);

  for (int pi = 0; pi < PPB_S; ++pi) {
    const int cur = pi & 1;
    const int p = p0 + pi;
    if (t < NPT) {                      // waves 0-1: stage + drain
      if (pi + 1 < PPB_S) {
        async_stage16(&fbuf[cur ^ 1][t][0],
                      feats + ((size_t)(p + 1) * NPT + t) * CIN);
        async_wait_le1();               // oldest (cur) load complete
      } else {
        async_wait_le0();
      }
    }
    __syncthreads();
    v8f c[4];
    pillar_compute(fbuf[cur], npts, coors, p, aug, sums, bfrag, c);
#pragma unroll
    for (int v = 0; v < 8; ++v) {
      sacc[v] += c[0][v] + c[1][v] + c[2][v] + c[3][v];
      qacc[v] += c[0][v] * c[0][v] + c[1][v] * c[1][v] +
                 c[2][v] * c[2][v] + c[3][v] * c[3][v];
    }
  }

  // One cross-lane reduction at the end (xor masks stay in 16-lane halves).
#pragma unroll
  for (int v = 0; v < 8; ++v) {
    float s = sacc[v], q = qacc[v];
#pragma unroll
    for (int m = 8; m > 0; m >>= 1) {
      s += __shfl_xor(s, m, 32);
      q += __shfl_xor(q, m, 32);
    }
    if ((lane & 15) == 0) {             // unique n per writer, no atomics
      const int n = (w << 4) + v + ((lane >> 4) << 3);
      ws[(size_t)blockIdx.x * 128 + n] = s;
      ws[(size_t)blockIdx.x * 128 + 64 + n] = q;
    }
  }
}

// ---- Pass 2: fold partials into affine BN params --------------------------
__global__ __launch_bounds__(64) void k_params(const float* __restrict__ gamma,
                                               const float* __restrict__ beta,
                                               float* __restrict__ ws) {
  const int n = threadIdx.x;  // 64 threads
  float S = 0.f, Q = 0.f;
  for (int b = 0; b < SBLK; ++b) {      // fixed order -> deterministic
    S += ws[(size_t)b * 128 + n];
    Q += ws[(size_t)b * 128 + 64 + n];
  }
  const float mean = S / COUNT_F;
  const float var = Q / COUNT_F - mean * mean;   // population variance
  const float a = rsqrtf(var + BN_EPS) * gamma[n];
  ws[PAR_OFF + n] = a;
  ws[PAR_OFF + 64 + n] = beta[n] - mean * a;
}

// ---- Pass 3: recompute GEMM, normalize, relu, max over n ------------------
__global__ __launch_bounds__(128) void k_out(const float* __restrict__ feats,
                                             const int* __restrict__ npts,
                                             const int* __restrict__ coors,
                                             const float* __restrict__ W,
                                             const float* __restrict__ ws,
                                             float* __restrict__ out) {
  __shared__ __align__(16) float fbuf[2][NPT][CIN];
  __shared__ float aug[NPT][KPAD];
  __shared__ float Wt[KPAD][OUTC];
  __shared__ float sums[4];
  __shared__ float la[NPT];
  __shared__ float lb[NPT];
  __shared__ float red[4][OUTC];

  const int t = threadIdx.x;
  const int lane = t & 31;
  const int w = t >> 5;
  if (t < NPT) {
    la[t] = ws[PAR_OFF + t];
    lb[t] = ws[PAR_OFF + 64 + t];
  }
  v2f bfrag[3][4];
  build_bfrags(W, Wt, bfrag);   // barrier inside also covers la/lb

  // Each lane's 8 accumulator slots map to 8 fixed n values; hoist params.
  float an[8], bn[8];
#pragma unroll
  for (int v = 0; v < 8; ++v) {
    const int n = (w << 4) + v + ((lane >> 4) << 3);
    an[v] = la[n];
    bn[v] = lb[n];
  }

  const int p0 = (int)blockIdx.x * PPB_O;
  if (t < NPT)
    async_stage16(&fbuf[0][t][0], feats + ((size_t)p0 * NPT + t) * CIN);

  for (int pi = 0; pi < PPB_O; ++pi) {
    const int cur = pi & 1;
    const int p = p0 + pi;
    if (t < NPT) {
      if (pi + 1 < PPB_O) {
        async_stage16(&fbuf[cur ^ 1][t][0],
                      feats + ((size_t)(p + 1) * NPT + t) * CIN);
        async_wait_le1();
      } else {
        async_wait_le0();
      }
    }
    __syncthreads();
    v8f c[4];
    pillar_compute(fbuf[cur], npts, coors, p, aug, sums, bfrag, c);
#pragma unroll
    for (int j = 0; j < 4; ++j) {
      float mx = -3.4e38f;
#pragma unroll
      for (int v = 0; v < 8; ++v) {
        mx = fmaxf(mx, an[v] * c[j][v] + bn[v]);
      }
      mx = fmaxf(mx, __shfl_xor(mx, 16, 32));   // merge the two n-halves
      if (lane < 16) red[w][j * 16 + lane] = mx;
    }
    __syncthreads();
    if (t < OUTC) {
      float r = fmaxf(fmaxf(red[0][t], red[1][t]),
                      fmaxf(red[2][t], red[3][t]));
      out[(size_t)p * OUTC + t] = fmaxf(r, 0.f);  // relu commutes with max
    }
  }
}

extern "C" void kernel_launch(void* const* d_in, const int* in_sizes, int n_in,
                              void* d_out, int out_size, void* d_ws,
                              size_t ws_size, hipStream_t stream) {
  const float* feats = (const float*)d_in[0];
  const int* npts    = (const int*)d_in[1];
  const int* coors   = (const int*)d_in[2];
  const float* W     = (const float*)d_in[3];
  const float* gamma = (const float*)d_in[4];
  const float* beta  = (const float*)d_in[5];
  float* ws  = (float*)d_ws;
  float* out = (float*)d_out;

  k_stats <<<SBLK, 128, 0, stream>>>(feats, npts, coors, W, ws);
  k_params<<<1, 64, 0, stream>>>(gamma, beta, ws);
  k_out   <<<OBLK, 128, 0, stream>>>(feats, npts, coors, W, ws, out);
}